// BS_Torch_55284819034371
// MI455X (gfx1250) — compile-verified
//
#include <hip/hip_runtime.h>

// ---------------------------------------------------------------------------
// Tricubic (order 3, degree p=2) tensor-product B-spline evaluation, 2M pts.
// Strategy:
//   1) prep kernel: transpose coeff (3,Nu,Nv,Nw) -> AoS float4 (Nu,Nv,Nw,4)
//      in d_ws so each of the 27 taps is one aligned b128 gather (L2 resident).
//   2) eval kernel: knots staged to LDS via gfx1250 global_load_async_to_lds,
//      arithmetic span search (+/-1 fixup vs the actual knot values to match
//      jnp.searchsorted), unrolled Cox-de-Boor, 27 float4 gathers, 3 outputs.
// ---------------------------------------------------------------------------

#define BS_EPS 1e-20f

// ---- gfx1250 async global->LDS staging (ASYNCcnt path) --------------------
__device__ __forceinline__ void async_g2l_b32(const float* gsrc, const float* ldst_generic) {
    // Low 32 bits of a generic pointer into the LDS aperture are the
    // wave-relative LDS byte offset (ISA 10.2).
    unsigned loff = (unsigned)(unsigned long long)ldst_generic;
    asm volatile("global_load_async_to_lds_b32 %0, %1, off"
                 :: "v"(loff), "v"(gsrc)
                 : "memory");
}

__device__ __forceinline__ void wait_async_zero() {
    asm volatile("s_wait_asynccnt 0" ::: "memory");
}

// ---- span search: smallest j>=1 with knots[2+j] >= X, idx = j-1 -----------
__device__ __forceinline__ int find_span(float X, const float* __restrict__ sk, int seg) {
    float t = X * (float)seg;
    int j = (int)t;                       // floor, t >= 0
    j = max(0, min(j, seg));
    if (sk[2 + j] >= X) {
        if (j > 0 && sk[1 + j] >= X) --j; // overshoot by one
    } else {
        ++j;                              // undershoot by one
    }
    j = max(1, min(j, seg));
    return j - 1;                         // idx in [0, seg-1]
}

// ---- degree-2 Cox-de-Boor, fully unrolled (matches reference loop) --------
__device__ __forceinline__ void basis2(float X, const float* __restrict__ kn, float* __restrict__ N) {
    float d32 = kn[3] - kn[2] + BS_EPS;
    float n1  = (kn[3] - X) / d32;        // level-1 basis
    float n2  = (X - kn[2]) / d32;
    float d31 = kn[3] - kn[1] + BS_EPS;
    float d42 = kn[4] - kn[2] + BS_EPS;
    N[0] = (kn[3] - X) / d31 * n1;
    N[1] = (X - kn[1]) / d31 * n1 + (kn[4] - X) / d42 * n2;
    N[2] = (X - kn[2]) / d42 * n2;
}

// ---- prep: (3,Nc) channel-major -> (Nc,4) float4 AoS ----------------------
__global__ void bs_transpose_coeff(const float* __restrict__ cf,
                                   float4* __restrict__ cfT, int Nc) {
    int i = blockIdx.x * blockDim.x + threadIdx.x;
    if (i < Nc) {
        float4 v;
        v.x = cf[i];
        v.y = cf[Nc + i];
        v.z = cf[2 * Nc + i];
        v.w = 0.0f;
        cfT[i] = v;
    }
}

// ---- main eval, AoS float4 gathers ----------------------------------------
__global__ void __launch_bounds__(256)
bs_eval_aos(const float* __restrict__ uvw,
            const float* __restrict__ knotx,
            const float* __restrict__ knoty,
            const float* __restrict__ knotz,
            const float4* __restrict__ cfT,
            float* __restrict__ out,
            int Np, int Nv, int Nw,
            int lenU, int lenV, int lenW,
            int segU, int segV, int segW) {
    __shared__ float sk[3 * 80];
    const int tid = threadIdx.x;

    // Stage 3 knot vectors to LDS with gfx1250 async copies.
    if (tid < lenU) async_g2l_b32(knotx + tid, &sk[tid]);
    if (tid < lenV) async_g2l_b32(knoty + tid, &sk[80 + tid]);
    if (tid < lenW) async_g2l_b32(knotz + tid, &sk[160 + tid]);
    wait_async_zero();
    __syncthreads();

    const int gid = blockIdx.x * blockDim.x + tid;
    if (gid >= Np) return;

    float u = fminf(fmaxf(uvw[gid],          1e-14f), 1.0f);
    float v = fminf(fmaxf(uvw[Np + gid],     1e-14f), 1.0f);
    float w = fminf(fmaxf(uvw[2 * Np + gid], 1e-14f), 1.0f);

    const int iu = find_span(u, sk,        segU);
    const int iv = find_span(v, sk + 80,   segV);
    const int iw = find_span(w, sk + 160,  segW);

    float NU[3], NV[3], NW[3];
    basis2(u, sk + iu,        NU);
    basis2(v, sk + 80 + iv,   NV);
    basis2(w, sk + 160 + iw,  NW);

    float ax = 0.f, ay = 0.f, az = 0.f;
    const float4* cbase = cfT + ((long)iu * Nv + iv) * Nw + iw;
#pragma unroll
    for (int ii = 0; ii < 3; ++ii) {
#pragma unroll
        for (int jj = 0; jj < 3; ++jj) {
            const float4* row = cbase + (ii * Nv + jj) * Nw;
            float wij = NU[ii] * NV[jj];
#pragma unroll
            for (int kk = 0; kk < 3; ++kk) {
                float4 c = row[kk];                 // aligned b128 gather, L2-hot
                float  wt = wij * NW[kk];
                ax = fmaf(c.x, wt, ax);
                ay = fmaf(c.y, wt, ay);
                az = fmaf(c.z, wt, az);
            }
        }
    }
    out[gid]          = ax;
    out[Np + gid]     = ay;
    out[2 * Np + gid] = az;
}

// ---- fallback: direct gathers from the original channel-major layout ------
__global__ void __launch_bounds__(256)
bs_eval_direct(const float* __restrict__ uvw,
               const float* __restrict__ knotx,
               const float* __restrict__ knoty,
               const float* __restrict__ knotz,
               const float* __restrict__ cf,
               float* __restrict__ out,
               int Np, int Nv, int Nw, int Nc,
               int lenU, int lenV, int lenW,
               int segU, int segV, int segW) {
    __shared__ float sk[3 * 80];
    const int tid = threadIdx.x;
    if (tid < lenU) async_g2l_b32(knotx + tid, &sk[tid]);
    if (tid < lenV) async_g2l_b32(knoty + tid, &sk[80 + tid]);
    if (tid < lenW) async_g2l_b32(knotz + tid, &sk[160 + tid]);
    wait_async_zero();
    __syncthreads();

    const int gid = blockIdx.x * blockDim.x + tid;
    if (gid >= Np) return;

    float u = fminf(fmaxf(uvw[gid],          1e-14f), 1.0f);
    float v = fminf(fmaxf(uvw[Np + gid],     1e-14f), 1.0f);
    float w = fminf(fmaxf(uvw[2 * Np + gid], 1e-14f), 1.0f);

    const int iu = find_span(u, sk,        segU);
    const int iv = find_span(v, sk + 80,   segV);
    const int iw = find_span(w, sk + 160,  segW);

    float NU[3], NV[3], NW[3];
    basis2(u, sk + iu,        NU);
    basis2(v, sk + 80 + iv,   NV);
    basis2(w, sk + 160 + iw,  NW);

    float ax = 0.f, ay = 0.f, az = 0.f;
    const long base = ((long)iu * Nv + iv) * Nw + iw;
#pragma unroll
    for (int ii = 0; ii < 3; ++ii) {
#pragma unroll
        for (int jj = 0; jj < 3; ++jj) {
            long rb = base + (long)(ii * Nv + jj) * Nw;
            float wij = NU[ii] * NV[jj];
#pragma unroll
            for (int kk = 0; kk < 3; ++kk) {
                float wt = wij * NW[kk];
                ax = fmaf(cf[rb + kk],          wt, ax);
                ay = fmaf(cf[Nc + rb + kk],     wt, ay);
                az = fmaf(cf[2 * Nc + rb + kk], wt, az);
            }
        }
    }
    out[gid]          = ax;
    out[Np + gid]     = ay;
    out[2 * Np + gid] = az;
}

extern "C" void kernel_launch(void* const* d_in, const int* in_sizes, int n_in,
                              void* d_out, int out_size, void* d_ws, size_t ws_size,
                              hipStream_t stream) {
    const float* uvw   = (const float*)d_in[0];
    const float* knotx = (const float*)d_in[1];
    const float* knoty = (const float*)d_in[2];
    const float* knotz = (const float*)d_in[3];
    const float* coeff = (const float*)d_in[4];
    // d_in[5] = order (always 3 per setup; degree p=2 is hardcoded)

    const int Np   = in_sizes[0] / 3;
    const int lenU = in_sizes[1], lenV = in_sizes[2], lenW = in_sizes[3];
    const int Nu   = lenU - 3, Nv = lenV - 3, Nw = lenW - 3;   // len = N + p + 1
    const int segU = Nu - 2, segV = Nv - 2, segW = Nw - 2;     // uniform interior spans
    const int Nc   = in_sizes[4] / 3;                          // Nu*Nv*Nw

    float* out = (float*)d_out;
    const int threads = 256;
    const int evalBlocks = (Np + threads - 1) / threads;

    if (ws_size >= (size_t)Nc * sizeof(float4)) {
        float4* cfT = (float4*)d_ws;
        bs_transpose_coeff<<<(Nc + threads - 1) / threads, threads, 0, stream>>>(coeff, cfT, Nc);
        bs_eval_aos<<<evalBlocks, threads, 0, stream>>>(
            uvw, knotx, knoty, knotz, cfT, out,
            Np, Nv, Nw, lenU, lenV, lenW, segU, segV, segW);
    } else {
        bs_eval_direct<<<evalBlocks, threads, 0, stream>>>(
            uvw, knotx, knoty, knotz, coeff, out,
            Np, Nv, Nw, Nc, lenU, lenV, lenW, segU, segV, segW);
    }
}